// ConsecutiveMatch_27599459844984
// MI455X (gfx1250) — compile-verified
//
#include <hip/hip_runtime.h>
#include <stdint.h>

typedef __attribute__((ext_vector_type(2))) float v2f;
typedef __attribute__((ext_vector_type(8))) float v8f;
typedef __attribute__((ext_vector_type(4))) int   v4i;

#define B_   8
#define N_   2048
#define M_   2048
#define D_   256
#define EPS_ 1e-8f

#define NT 64          // src rows per block
#define MT 256         // dst cols per chunk (block column tile)
#define KC 32          // k-chunk staged for B
#define A_STRIDE 260   // 256 + 4 pad -> conflict-free 16x4 frag reads
#define B_STRIDE 36    // 32 + 4 pad  -> conflict-free 16x4 frag reads
#define NCHUNK ((M_ / MT) * (D_ / KC))   // 64 pipelined B chunks

#if defined(__has_builtin)
#  if __has_builtin(__builtin_amdgcn_global_load_async_to_lds_b128)
#    define HAVE_ASYNC_LDS 1
#  endif
#endif
#ifndef HAVE_ASYNC_LDS
#  define HAVE_ASYNC_LDS 0
#endif

__device__ __forceinline__ void wait_async0() {
#if HAVE_ASYNC_LDS
#  if __has_builtin(__builtin_amdgcn_s_wait_asynccnt)
    __builtin_amdgcn_s_wait_asynccnt(0);
#  else
    asm volatile("s_wait_asynccnt 0x0" ::: "memory");
#  endif
#endif
}

// Copy one 16-byte quantum global -> LDS (async DMA if available).
__device__ __forceinline__ void cp16(const float* __restrict__ g, float* l) {
#if HAVE_ASYNC_LDS
    __builtin_amdgcn_global_load_async_to_lds_b128(
        (__attribute__((address_space(1))) v4i*)g,
        (__attribute__((address_space(3))) v4i*)l,
        0, 0);
#else
    *(float4*)l = *(const float4*)g;
#endif
}

// ---------------------------------------------------------------------------
// Kernel 1: row L2 norms for desc_src (B*N rows) then desc_dst (B*M rows).
// ---------------------------------------------------------------------------
__global__ void cm_norms_kernel(const float* __restrict__ src,
                                const float* __restrict__ dst,
                                float* __restrict__ ws_norm) {
    const int wid  = threadIdx.x >> 5;
    const int lane = threadIdx.x & 31;
    const int gr   = blockIdx.x * 8 + wid;          // 0 .. B*(N+M)-1

    const float* base = (gr < B_ * N_) ? (src + (size_t)gr * D_)
                                       : (dst + (size_t)(gr - B_ * N_) * D_);
    const float4* p = (const float4*)base;          // 64 float4 per row
    float4 x0 = p[lane];
    float4 x1 = p[lane + 32];
    float s = x0.x * x0.x + x0.y * x0.y + x0.z * x0.z + x0.w * x0.w +
              x1.x * x1.x + x1.y * x1.y + x1.z * x1.z + x1.w * x1.w;
#pragma unroll
    for (int m = 16; m >= 1; m >>= 1) s += __shfl_xor(s, m, 32);
    if (lane == 0) ws_norm[gr] = sqrtf(s);
}

// ---------------------------------------------------------------------------
// Kernel 2: per block: 64 src rows x full M sweep.
//   - A slab (64x256 f32) resident in LDS, loaded via async DMA.
//   - B streamed in 256x32 chunks, double-buffered async global->LDS.
//   - 8 waves (2 row-groups x 4 col-groups), each wave owns 2x4 WMMA C tiles.
//   - fused scale + running max/argmax via packed-u64 LDS atomicMax.
// ---------------------------------------------------------------------------
__global__ __launch_bounds__(256)
void cm_match_kernel(const float* __restrict__ src,
                     const float* __restrict__ dst,
                     const float* __restrict__ pts,
                     const float* __restrict__ ws_nsrc,
                     const float* __restrict__ ws_ndst,
                     float* __restrict__ out_matched,   // [B,N,2]
                     float* __restrict__ out_conf) {    // [B,N]
    __shared__ __align__(16) float lds_a[NT * A_STRIDE];       // ~65 KB
    __shared__ __align__(16) float lds_b[2][MT * B_STRIDE];    // ~72 KB
    __shared__ float ns_row[NT];
    __shared__ float nd_col[MT];
    __shared__ unsigned long long best[NT];

    const int b    = blockIdx.x / (N_ / NT);
    const int n0   = (blockIdx.x % (N_ / NT)) * NT;
    const int tid  = threadIdx.x;
    const int lane = tid & 31;
    const int wid  = tid >> 5;
    const int wr   = wid >> 2;        // 0..1 : 32-row group
    const int wc   = wid & 3;         // 0..3 : 64-col group
    const int hl   = lane >> 4;       // lane half (K select / C row select)
    const int ll   = lane & 15;       // M/N index inside 16-wide tile

    // ---- prologue: async-stage A slab + B chunk 0, init tables ----
    const float* gsrc = src + ((size_t)b * N_ + n0) * D_;
    for (int i = tid; i < NT * (D_ / 4); i += 256) {        // 16 per thread
        const int row = i >> 6, kf = i & 63;
        cp16(gsrc + row * D_ + kf * 4, &lds_a[row * A_STRIDE + kf * 4]);
    }
#pragma unroll
    for (int j = 0; j < 8; ++j) {                           // B chunk 0 -> buf 0
        const int i = tid + j * 256;
        const int col = i >> 3, kf = i & 7;
        cp16(dst + ((size_t)(b * M_ + col)) * D_ + kf * 4,
             &lds_b[0][col * B_STRIDE + kf * 4]);
    }
    if (tid < NT) {
        best[tid]   = 0ull;
        ns_row[tid] = ws_nsrc[b * N_ + n0 + tid];
    }
    wait_async0();
    __syncthreads();

    for (int mt = 0; mt < M_ / MT; ++mt) {
        const int m0 = mt * MT;
        nd_col[tid] = ws_ndst[b * M_ + m0 + tid];           // MT == blockDim
        v8f acc[2][4] = {};

        for (int kc = 0; kc < D_ / KC; ++kc) {
            const int c  = mt * (D_ / KC) + kc;             // flat chunk id
            const int cn = c + 1;
            const int cur = c & 1;

            // issue async prefetch of next chunk into the other buffer
            if (cn < NCHUNK) {
                const int m0n = (cn >> 3) * MT, k0n = (cn & 7) * KC;
#pragma unroll
                for (int j = 0; j < 8; ++j) {
                    const int i = tid + j * 256;
                    const int col = i >> 3, kf = i & 7;
                    cp16(dst + ((size_t)(b * M_ + m0n + col)) * D_ + k0n + kf * 4,
                         &lds_b[cn & 1][col * B_STRIDE + kf * 4]);
                }
            }

            // compute on current buffer: 8 K-steps of 4, 8 WMMAs per step
            const float* lb = lds_b[cur];
#pragma unroll
            for (int ks = 0; ks < KC / 4; ++ks) {
                const int kofs = ks * 4 + 2 * hl;           // frag k base
                v2f af[2], bf[4];
#pragma unroll
                for (int ri = 0; ri < 2; ++ri)
                    af[ri] = *(const v2f*)&lds_a[(wr * 32 + ri * 16 + ll) * A_STRIDE
                                                 + kc * KC + kofs];
#pragma unroll
                for (int ci = 0; ci < 4; ++ci)
                    bf[ci] = *(const v2f*)&lb[(wc * 64 + ci * 16 + ll) * B_STRIDE
                                              + kofs];
#pragma unroll
                for (int ri = 0; ri < 2; ++ri)
#pragma unroll
                    for (int ci = 0; ci < 4; ++ci)
                        acc[ri][ci] = __builtin_amdgcn_wmma_f32_16x16x4_f32(
                            false, af[ri], false, bf[ci],
                            (short)0, acc[ri][ci], false, false);
            }

            // chunk boundary: my prefetch done + everyone done reading
            if (cn < NCHUNK) {
                wait_async0();
                __syncthreads();
            }
        }

        // scale by 1/max(|x||y|, eps); fold into running max/argmax
#pragma unroll
        for (int ri = 0; ri < 2; ++ri) {
#pragma unroll
            for (int v = 0; v < 8; ++v) {
                const int rowl = wr * 32 + ri * 16 + v + 8 * hl;
                const float ns = ns_row[rowl];
                unsigned long long pk = 0ull;
#pragma unroll
                for (int ci = 0; ci < 4; ++ci) {
                    const int coll = wc * 64 + ci * 16 + ll;
                    const float sim = acc[ri][ci][v] / fmaxf(ns * nd_col[coll], EPS_);
                    unsigned u = __float_as_uint(sim);
                    u = (u & 0x80000000u) ? ~u : (u | 0x80000000u); // order-preserving
                    const unsigned long long cand =
                        ((unsigned long long)u << 32) | (unsigned)(~(m0 + coll));
                    pk = (cand > pk) ? cand : pk;
                }
                // reduce across the 16 lanes sharing this row (within half)
                unsigned hi = (unsigned)(pk >> 32), lo = (unsigned)pk;
#pragma unroll
                for (int m = 8; m >= 1; m >>= 1) {
                    const unsigned ohi = __shfl_xor(hi, m, 16);
                    const unsigned olo = __shfl_xor(lo, m, 16);
                    const unsigned long long o =
                        ((unsigned long long)ohi << 32) | olo;
                    if (o > pk) { pk = o; hi = ohi; lo = olo; }
                }
                if (ll == 0) atomicMax(&best[rowl], pk);    // ds_max_u64
            }
        }
        __syncthreads();   // nd_col rewrite next mt must not race scale reads
    }

    // Decode: confidence + gather matched points.
    if (tid < NT) {
        const unsigned long long pk = best[tid];
        const unsigned key = (unsigned)(pk >> 32);
        const unsigned ub  = (key & 0x80000000u) ? (key & 0x7FFFFFFFu) : ~key;
        const int col  = (int)(~(unsigned)pk);
        const int grow = b * N_ + n0 + tid;
        out_conf[grow] = __uint_as_float(ub);
        const float2 p = *(const float2*)(pts + ((size_t)b * M_ + col) * 2);
        ((float2*)out_matched)[grow] = p;
    }
}

// ---------------------------------------------------------------------------
extern "C" void kernel_launch(void* const* d_in, const int* in_sizes, int n_in,
                              void* d_out, int out_size, void* d_ws, size_t ws_size,
                              hipStream_t stream) {
    const float* desc_src   = (const float*)d_in[0];   // [B,N,D]
    const float* desc_dst   = (const float*)d_in[1];   // [B,M,D]
    const float* points_dst = (const float*)d_in[2];   // [B,M,2]

    float* out_matched = (float*)d_out;                        // [B,N,2] first
    float* out_conf    = (float*)d_out + (size_t)B_ * N_ * 2;  // then [B,N]

    float* ws_norm = (float*)d_ws;                     // B*N + B*M floats
    float* ws_nsrc = ws_norm;
    float* ws_ndst = ws_norm + (size_t)B_ * N_;

    cm_norms_kernel<<<(B_ * (N_ + M_)) / 8, 256, 0, stream>>>(
        desc_src, desc_dst, ws_norm);

    cm_match_kernel<<<B_ * (N_ / NT), 256, 0, stream>>>(
        desc_src, desc_dst, points_dst, ws_nsrc, ws_ndst,
        out_matched, out_conf);
}